// IouRegLoss_28363964023473
// MI455X (gfx1250) — compile-verified
//
#include <hip/hip_runtime.h>

// IoU regression loss (CenterPoint-style) for MI455X / gfx1250, wave32.
//
// Shapes: box_pred [16, 214272, 7] f32, mask [16,500] i32, ind [16,500] i32,
//         box_gt [16,500,7] f32  ->  scalar f32.
//
// Memory-bound analysis: ~0.5 MB touched total -> launch-latency bound.
// One workgroup (32 wave32s), deterministic reduction:
//   per-thread accumulate -> per-wave WMMA f32 16x16x4 double-reduction
//   -> 32 wave partials in LDS -> serial sum by thread 0.

typedef __attribute__((ext_vector_type(2))) float v2f;
typedef __attribute__((ext_vector_type(8))) float v8f;

#define B_     16
#define HW_    214272
#define C_     7
#define M_     500
#define TOTAL  (B_ * M_)      // 8000
#define NTH    1024           // 32 waves of 32
#define NWAVE  (NTH / 32)

// Sum a per-lane float across all 32 lanes of a wave using two chained
// V_WMMA_F32_16X16X4_F32 ops (full f32 precision).
//
// Step 1: A = ones(16x4), B holds the 32 partials (one per lane in B-VGPR0,
//         zeros in B-VGPR1), C = 0.
//         => D1[m][n] = sum_k B[k][n]  (column sums, replicated in every row,
//            independent of B's exact lane layout).
// Step 2: A2 = D1's VGPR0/1. Per the documented A layout (lane L<16 supplies
//         row M=L for K=0,1; lane L>=16 supplies row M=L-16 for K=2,3) and the
//         documented D layout (lane L<16 VGPRv = D[v][L], lane L>=16 VGPRv =
//         D[v+8][L-16]), this yields A2[m][k] = colsum(m) for all k.
//         With B = ones: D2[m][n] = 4*colsum(m).
//         Lanes 0..15 hold rows 0..7, lanes 16..31 hold rows 8..15, so
//         sum8(d2) + shfl_xor(.,16) = 4 * total.
__device__ __forceinline__ float wave_sum_wmma(float p) {
    v2f ones; ones[0] = 1.0f; ones[1] = 1.0f;
    v2f bmat; bmat[0] = p;    bmat[1] = 0.0f;
    v8f zero = {};
    // 8-arg f32 WMMA: (neg_a, A, neg_b, B, c_mod, C, reuse_a, reuse_b)
    v8f d1 = __builtin_amdgcn_wmma_f32_16x16x4_f32(
        false, ones, false, bmat, (short)0, zero, false, false);
    v2f a2; a2[0] = d1[0]; a2[1] = d1[1];
    v8f d2 = __builtin_amdgcn_wmma_f32_16x16x4_f32(
        false, a2, false, ones, (short)0, zero, false, false);
    float t = ((d2[0] + d2[1]) + (d2[2] + d2[3])) +
              ((d2[4] + d2[5]) + (d2[6] + d2[7]));
    t += __shfl_xor(t, 16, 32);
    return t * 0.25f;
}

__global__ __launch_bounds__(NTH)
void iou_reg_loss_kernel(const float* __restrict__ box_pred,
                         const int*   __restrict__ mask,
                         const int*   __restrict__ ind,
                         const float* __restrict__ box_gt,
                         float*       __restrict__ out) {
    __shared__ float s_num[NWAVE];
    __shared__ float s_den[NWAVE];

    const int tid = threadIdx.x;
    float num = 0.0f;   // sum of m * (1 - iou)
    float den = 0.0f;   // sum of m

    for (int i = tid; i < TOTAL; i += NTH) {
        const int mk = mask[i];
        if (mk != 0) {  // zero-mask terms contribute exactly 0 -> skip gather
            const int   b   = i / M_;
            const int   idx = ind[i];
            const float fm  = (float)mk;

            // Sparse gather from the 96MB pred tensor: non-temporal,
            // scalar dword loads (boxes are only 4B aligned, stride 28B).
            const float* p = box_pred + ((size_t)b * HW_ + (size_t)idx) * C_;
            const float* g = box_gt + (size_t)i * C_;
            const float px = __builtin_nontemporal_load(p + 0);
            const float py = __builtin_nontemporal_load(p + 1);
            const float pz = __builtin_nontemporal_load(p + 2);
            const float pw = __builtin_nontemporal_load(p + 3);
            const float pl = __builtin_nontemporal_load(p + 4);
            const float ph = __builtin_nontemporal_load(p + 5);
            const float gx = g[0], gy = g[1], gz = g[2];
            const float gw = g[3], gl = g[4], gh = g[5];

            const float lx = fmaxf(px - pw * 0.5f, gx - gw * 0.5f);
            const float ly = fmaxf(py - pl * 0.5f, gy - gl * 0.5f);
            const float lz = fmaxf(pz - ph * 0.5f, gz - gh * 0.5f);
            const float rx = fminf(px + pw * 0.5f, gx + gw * 0.5f);
            const float ry = fminf(py + pl * 0.5f, gy + gl * 0.5f);
            const float rz = fminf(pz + ph * 0.5f, gz + gh * 0.5f);

            const float iw = fmaxf(rx - lx, 0.0f);
            const float ih = fmaxf(ry - ly, 0.0f);
            const float id = fmaxf(rz - lz, 0.0f);
            const float inter = iw * ih * id;
            const float vol_p = pw * pl * ph;
            const float vol_g = gw * gl * gh;
            const float uni   = vol_p + vol_g - inter;
            const float iou   = inter / (uni + 1e-7f);

            num += fm * (1.0f - iou);
            den += fm;
        }
    }

    // Per-wave WMMA reduction. Called unconditionally by every thread so
    // EXEC is all-ones inside each wave (WMMA requirement).
    const float wnum = wave_sum_wmma(num);
    const float wden = wave_sum_wmma(den);

    const int wave = tid >> 5;
    const int lane = tid & 31;
    if (lane == 0) {
        s_num[wave] = wnum;
        s_den[wave] = wden;
    }
    __syncthreads();

    if (tid == 0) {
        float tn = 0.0f, td = 0.0f;
#pragma unroll
        for (int w = 0; w < NWAVE; ++w) {
            tn += s_num[w];
            td += s_den[w];
        }
        out[0] = tn / (td + 1e-4f);
    }
}

extern "C" void kernel_launch(void* const* d_in, const int* in_sizes, int n_in,
                              void* d_out, int out_size, void* d_ws, size_t ws_size,
                              hipStream_t stream) {
    (void)in_sizes; (void)n_in; (void)out_size; (void)d_ws; (void)ws_size;
    const float* box_pred = (const float*)d_in[0];
    const int*   mask     = (const int*)d_in[1];
    const int*   ind      = (const int*)d_in[2];
    const float* box_gt   = (const float*)d_in[3];
    float*       out      = (float*)d_out;

    iou_reg_loss_kernel<<<dim3(1), dim3(NTH), 0, stream>>>(
        box_pred, mask, ind, box_gt, out);
}